// NeuralODE_58342835749494
// MI455X (gfx1250) — compile-verified
//
#include <hip/hip_runtime.h>
#include <math.h>

// Problem constants (match reference)
#define B_SZ    256
#define ZD      1024
#define HID     2048
#define T_STEPS 128
#define DT_VAL  (1.0f / 127.0f)

// GEMM tiling
#define KC      32      // K-chunk staged in LDS
#define KC_PAD  36      // padded LDS row stride (144B rows: 16B aligned, bank-spread)
#define BM      64      // block tile M (2 waves)
#define BN      128     // block tile N (4 waves)
// per-wave tile: 32x32 = 2x2 WMMA 16x16 tiles; 256 threads = 8 waves

typedef float v2f __attribute__((ext_vector_type(2)));
typedef float v8f __attribute__((ext_vector_type(8)));

// ---- CDNA5 async global->LDS copy (ASYNCcnt-tracked, bypasses VGPR staging)
#define ASYNC_LOAD_B128(lds_off, gaddr)                                  \
    asm volatile("global_load_async_to_lds_b128 %0, %1, off"             \
                 :: "v"(lds_off), "v"(gaddr) : "memory")
#define WAIT_ASYNCCNT(n)                                                 \
    asm volatile("s_wait_asynccnt " #n ::: "memory")

__device__ __forceinline__ unsigned lds_off_of(const void* p) {
    // flat shared pointer: low 32 bits are the LDS offset (ISA 10.2)
    return (unsigned)(uintptr_t)p;
}

// Issue one K-chunk's worth of async copies: A tile 64xKC (2 b128/thread),
// B tile 128xKC (4 b128/thread). 6 outstanding async ops per thread.
__device__ __forceinline__ void issue_chunk(
    const float* __restrict__ A, const float* __restrict__ Bw, int K,
    int mBase, int nBase, int k0, float* At, float* Bt, int tid)
{
    #pragma unroll
    for (int j = 0; j < 2; ++j) {
        int f  = tid + j * 256;
        int r  = f >> 3;
        int c4 = (f & 7) << 2;
        unsigned           lo = lds_off_of(&At[r * KC_PAD + c4]);
        unsigned long long ga =
            (unsigned long long)(uintptr_t)(A + (size_t)(mBase + r) * K + k0 + c4);
        ASYNC_LOAD_B128(lo, ga);
    }
    #pragma unroll
    for (int j = 0; j < 4; ++j) {
        int f  = tid + j * 256;
        int r  = f >> 3;
        int c4 = (f & 7) << 2;
        unsigned           lo = lds_off_of(&Bt[r * KC_PAD + c4]);
        unsigned long long ga =
            (unsigned long long)(uintptr_t)(Bw + (size_t)(nBase + r) * K + k0 + c4);
        ASYNC_LOAD_B128(lo, ga);
    }
}

// Shared double-buffered fp32 WMMA mainloop: acc[2][2] (+) A[64xK] * B[128xK]^T
__device__ __forceinline__ void gemm_mainloop(
    const float* __restrict__ A, const float* __restrict__ Bw, int K,
    int mBase, int nBase, float* AtBuf, float* BtBuf, v8f acc[2][2])
{
    const int tid  = threadIdx.x;
    const int lane = tid & 31;
    const int wv   = tid >> 5;
    const int lm   = lane & 15;
    const int lh   = lane >> 4;
    const int wm   = wv >> 2;     // 0..1
    const int wn   = wv & 3;      // 0..3

    const int nChunks = K / KC;

    // Prologue: stage chunk 0 into buffer 0
    issue_chunk(A, Bw, K, mBase, nBase, 0, AtBuf, BtBuf, tid);

    for (int ch = 0; ch < nChunks; ++ch) {
        const int cur = ch & 1;
        float* Atc = AtBuf + cur * (BM * KC_PAD);
        float* Btc = BtBuf + cur * (BN * KC_PAD);

        if (ch + 1 < nChunks) {
            // Prefetch next chunk into the other buffer, then wait for the
            // previous 6 copies (in-order completion => <=6 outstanding).
            issue_chunk(A, Bw, K, mBase, nBase, (ch + 1) * KC,
                        AtBuf + (cur ^ 1) * (BM * KC_PAD),
                        BtBuf + (cur ^ 1) * (BN * KC_PAD), tid);
            WAIT_ASYNCCNT(6);
        } else {
            WAIT_ASYNCCNT(0);
        }
        __syncthreads();

        #pragma unroll
        for (int kk = 0; kk < KC; kk += 4) {
            v2f a[2], b[2];
            #pragma unroll
            for (int mt = 0; mt < 2; ++mt)
                a[mt] = *(const v2f*)&Atc[(wm * 32 + mt * 16 + lm) * KC_PAD + kk + lh * 2];
            #pragma unroll
            for (int nt = 0; nt < 2; ++nt)
                b[nt] = *(const v2f*)&Btc[(wn * 32 + nt * 16 + lm) * KC_PAD + kk + lh * 2];
            #pragma unroll
            for (int mt = 0; mt < 2; ++mt)
                #pragma unroll
                for (int nt = 0; nt < 2; ++nt)
                    acc[mt][nt] = __builtin_amdgcn_wmma_f32_16x16x4_f32(
                        false, a[mt], false, b[nt], (short)0, acc[mt][nt], false, false);
        }
        __syncthreads();
    }
}

// ---------------------------------------------------------------------------
// GEMM1: H[B,HID] = tanh(Z[B,ZD] * W1[HID,ZD]^T + t*wt + b1)
// ---------------------------------------------------------------------------
__global__ __launch_bounds__(256) void node_gemm1(
    const float* __restrict__ Z, const float* __restrict__ W1,
    const float* __restrict__ wt, const float* __restrict__ b1,
    float* __restrict__ H, float t)
{
    __shared__ float At[2 * BM * KC_PAD];
    __shared__ float Bt[2 * BN * KC_PAD];

    const int lane = threadIdx.x & 31;
    const int wv   = threadIdx.x >> 5;
    const int lm   = lane & 15;
    const int lh   = lane >> 4;
    const int wm   = wv >> 2;
    const int wn   = wv & 3;

    const int mBase = blockIdx.y * BM;
    const int nBase = blockIdx.x * BN;

    v8f acc[2][2];
    #pragma unroll
    for (int i = 0; i < 2; ++i)
        #pragma unroll
        for (int j = 0; j < 2; ++j)
            acc[i][j] = {};

    gemm_mainloop(Z, W1, ZD, mBase, nBase, At, Bt, acc);

    #pragma unroll
    for (int mt = 0; mt < 2; ++mt) {
        #pragma unroll
        for (int nt = 0; nt < 2; ++nt) {
            int n = nBase + wn * 32 + nt * 16 + lm;
            float add = t * wt[n] + b1[n];
            #pragma unroll
            for (int r = 0; r < 8; ++r) {
                int m = mBase + wm * 32 + mt * 16 + lh * 8 + r;
                H[(size_t)m * HID + n] = tanhf(acc[mt][nt][r] + add);
            }
        }
    }
}

// ---------------------------------------------------------------------------
// GEMM2: z_next = z + dt * (H[B,HID] * W2[ZD,HID]^T + b2); write traj column
// ---------------------------------------------------------------------------
__global__ __launch_bounds__(256) void node_gemm2(
    const float* __restrict__ Hm, const float* __restrict__ W2,
    const float* __restrict__ b2, const float* __restrict__ zin,
    float* __restrict__ zout, float* __restrict__ traj, int step)
{
    __shared__ float At[2 * BM * KC_PAD];
    __shared__ float Bt[2 * BN * KC_PAD];

    const int lane = threadIdx.x & 31;
    const int wv   = threadIdx.x >> 5;
    const int lm   = lane & 15;
    const int lh   = lane >> 4;
    const int wm   = wv >> 2;
    const int wn   = wv & 3;

    const int mBase = blockIdx.y * BM;
    const int nBase = blockIdx.x * BN;

    v8f acc[2][2];
    #pragma unroll
    for (int i = 0; i < 2; ++i)
        #pragma unroll
        for (int j = 0; j < 2; ++j)
            acc[i][j] = {};

    gemm_mainloop(Hm, W2, HID, mBase, nBase, At, Bt, acc);

    #pragma unroll
    for (int mt = 0; mt < 2; ++mt) {
        #pragma unroll
        for (int nt = 0; nt < 2; ++nt) {
            int z = nBase + wn * 32 + nt * 16 + lm;
            float bb = b2[z];
            #pragma unroll
            for (int r = 0; r < 8; ++r) {
                int bRow = mBase + wm * 32 + mt * 16 + lh * 8 + r;
                float zn = zin[(size_t)bRow * ZD + z] + DT_VAL * (acc[mt][nt][r] + bb);
                zout[(size_t)bRow * ZD + z] = zn;
                traj[(size_t)bRow * ZD * T_STEPS + (size_t)z * T_STEPS + (step + 1)] = zn;
            }
        }
    }
}

// traj[:, :, 0] = z0
__global__ __launch_bounds__(256) void node_init(const float* __restrict__ z0,
                                                 float* __restrict__ traj)
{
    int idx = blockIdx.x * 256 + threadIdx.x;   // over B*ZD
    int b = idx / ZD;
    int z = idx % ZD;
    traj[(size_t)b * ZD * T_STEPS + (size_t)z * T_STEPS] = z0[idx];
}

extern "C" void kernel_launch(void* const* d_in, const int* in_sizes, int n_in,
                              void* d_out, int out_size, void* d_ws, size_t ws_size,
                              hipStream_t stream)
{
    const float* z0 = (const float*)d_in[0];
    // d_in[1] = ts (uniform grid; t recomputed on host as s*DT)
    const float* W1 = (const float*)d_in[2];
    const float* wt = (const float*)d_in[3];
    const float* b1 = (const float*)d_in[4];
    const float* W2 = (const float*)d_in[5];
    const float* b2 = (const float*)d_in[6];
    float* traj = (float*)d_out;

    char* ws = (char*)d_ws;
    float* Hbuf  = (float*)ws;                                        // 2 MB
    float* zbuf0 = (float*)(ws + (size_t)B_SZ * HID * sizeof(float)); // 1 MB
    float* zbuf1 = zbuf0 + (size_t)B_SZ * ZD;                         // 1 MB

    dim3 blk(256);

    node_init<<<dim3((B_SZ * ZD) / 256), blk, 0, stream>>>(z0, traj);

    const float* zin = z0;
    for (int s = 0; s < T_STEPS - 1; ++s) {
        float t = (float)s * DT_VAL;
        float* zo = (s & 1) ? zbuf1 : zbuf0;
        node_gemm1<<<dim3(HID / BN, B_SZ / BM), blk, 0, stream>>>(zin, W1, wt, b1, Hbuf, t);
        node_gemm2<<<dim3(ZD / BN, B_SZ / BM), blk, 0, stream>>>(Hbuf, W2, b2, zin, zo, traj, s);
        zin = zo;
    }
}